// GCNWithJK_49718541418974
// MI455X (gfx1250) — compile-verified
//
#include <hip/hip_runtime.h>
#include <math.h>

#define N_NODES 50000
#define N_EDGES 800000
#define F_IN    128
#define HID     64
#define N_GRAPHS 64
#define N_CLASS 2
#define JK      (3 * HID)    // 192 (JK concat width)
#define GFEAT   (3 * JK)     // 576 (add|mean|max concat)

typedef __attribute__((ext_vector_type(2))) float v2f;
typedef __attribute__((ext_vector_type(8))) float v8f;

// ---------------------------------------------------------------------------
// Degree / normalization kernels
// ---------------------------------------------------------------------------
__global__ void zero_f32(float* __restrict__ p, int n) {
    int i = blockIdx.x * blockDim.x + threadIdx.x;
    if (i < n) p[i] = 0.0f;
}

__global__ void deg_accum(const int* __restrict__ dst, float* __restrict__ deg, int E) {
    int e = blockIdx.x * blockDim.x + threadIdx.x;
    if (e < E) atomicAdd(&deg[dst[e]], 1.0f);
}

// in-place: deg -> dinv = rsqrt(deg + 1)
__global__ void make_dinv(float* __restrict__ d, int n) {
    int i = blockIdx.x * blockDim.x + threadIdx.x;
    if (i < n) d[i] = rsqrtf(d[i] + 1.0f);
}

__global__ void make_enorm(const int* __restrict__ src, const int* __restrict__ dst,
                           const float* __restrict__ dinv, float* __restrict__ enorm, int E) {
    int e = blockIdx.x * blockDim.x + threadIdx.x;
    if (e < E) enorm[e] = dinv[src[e]] * dinv[dst[e]];
}

// ---------------------------------------------------------------------------
// WMMA f32 GEMM:  C[N x 64] = A[N x K] (row-major, stride lda) @ B[K x 64]
// One block = 128 threads = 4 waves; wave w computes a 16x16 tile at cols w*16.
// Block b computes rows [16b, 16b+16). N must be a multiple of 16.
// A-operand ISA layout (16x4 f32): lanes 0-15 hold K=k0,k0+1; lanes 16-31 hold
// K=k0+2,k0+3 (for their same M rows). B mirrored with N=lane%16.
// D layout: VGPR j -> row (j + 8*(lane>=16)), col = lane%16.
// ---------------------------------------------------------------------------
__global__ __launch_bounds__(128)
void gemm_wmma_f32(const float* __restrict__ A, const float* __restrict__ B,
                   float* __restrict__ C, int N, int K, int lda, int ldc) {
    const int wave = threadIdx.x >> 5;      // 0..3 -> column tile
    const int lane = threadIdx.x & 31;
    const int half = lane >> 4;             // 0 or 1
    const int r    = lane & 15;
    const int row0 = blockIdx.x * 16;
    const int col0 = wave * 16;
    if (row0 >= N) return;                  // uniform per block (EXEC stays all-1s)

    v8f acc = {};
    const float* __restrict__ arow = A + (size_t)(row0 + r) * lda;
    for (int k0 = 0; k0 < K; k0 += 4) {
        const int ka = k0 + 2 * half;
        v2f a, b;
        a.x = arow[ka];
        a.y = arow[ka + 1];
        b.x = B[(size_t)ka * HID + col0 + r];
        b.y = B[(size_t)(ka + 1) * HID + col0 + r];
        acc = __builtin_amdgcn_wmma_f32_16x16x4_f32(
            /*neg_a=*/false, a, /*neg_b=*/false, b,
            /*c_mod=*/(short)0, acc, /*reuse_a=*/false, /*reuse_b=*/false);
    }
#pragma unroll
    for (int j = 0; j < 8; ++j)
        C[(size_t)(row0 + j + 8 * half) * ldc + col0 + r] = acc[j];
}

// ---------------------------------------------------------------------------
// GCN epilogue kernels (out slice lives inside hcat, stride JK=192)
// ---------------------------------------------------------------------------
// out[n,f] = dinv[n]^2 * h[n,f] + bias[f]
__global__ void init_out(const float* __restrict__ h, const float* __restrict__ dinv,
                         const float* __restrict__ bias, float* __restrict__ out) {
    int i = blockIdx.x * blockDim.x + threadIdx.x;
    if (i >= N_NODES * HID) return;
    int n = i >> 6, f = i & (HID - 1);
    float dv = dinv[n];
    out[(size_t)n * JK + f] = dv * dv * h[i] + bias[f];
}

// out[dst,f] += h[src,f] * enorm[e]
__global__ void scatter_edges(const int* __restrict__ src, const int* __restrict__ dst,
                              const float* __restrict__ enorm, const float* __restrict__ h,
                              float* __restrict__ out) {
    long long i = (long long)blockIdx.x * blockDim.x + threadIdx.x;
    if (i >= (long long)N_EDGES * HID) return;
    int e = (int)(i >> 6), f = (int)(i & (HID - 1));
    float v = h[(size_t)src[e] * HID + f] * enorm[e];
    atomicAdd(&out[(size_t)dst[e] * JK + f], v);
}

__global__ void relu_slice(float* __restrict__ out) {
    int i = blockIdx.x * blockDim.x + threadIdx.x;
    if (i >= N_NODES * HID) return;
    int n = i >> 6, f = i & (HID - 1);
    size_t idx = (size_t)n * JK + f;
    out[idx] = fmaxf(out[idx], 0.0f);
}

// ---------------------------------------------------------------------------
// Graph pooling: batch[i] = (i*G)//N is sorted -> graph g owns node range
// [ceil(g*N/G), ceil((g+1)*N/G)).  grid = 64 graphs, block = 192 features.
// g_out[g] = [sum(192) | mean(192) | max(192)]
// ---------------------------------------------------------------------------
__global__ __launch_bounds__(JK)
void pool_kernel(const float* __restrict__ hcat, float* __restrict__ g_out) {
    int g = blockIdx.x;
    int f = threadIdx.x;  // 0..191
    int start = (g * N_NODES + N_GRAPHS - 1) / N_GRAPHS;
    int end   = ((g + 1) * N_NODES + N_GRAPHS - 1) / N_GRAPHS;
    float sum = 0.0f, mx = -INFINITY;
    for (int n = start; n < end; ++n) {
        float v = hcat[(size_t)n * JK + f];
        sum += v;
        mx = fmaxf(mx, v);
    }
    float cnt = (float)(end - start);
    g_out[g * GFEAT + f]          = sum;
    g_out[g * GFEAT + JK + f]     = sum / cnt;
    g_out[g * GFEAT + 2 * JK + f] = mx;
}

// s[i] = relu(gt[i] + b[i % 64])
__global__ void bias_relu(const float* __restrict__ gt, const float* __restrict__ b,
                          float* __restrict__ s, int n) {
    int i = blockIdx.x * blockDim.x + threadIdx.x;
    if (i < n) s[i] = fmaxf(gt[i] + b[i & (HID - 1)], 0.0f);
}

// logits = s @ lin2_W + lin2_b ; out = log_softmax(logits)
__global__ void head_kernel(const float* __restrict__ s, const float* __restrict__ W,
                            const float* __restrict__ b, float* __restrict__ out) {
    int g = blockIdx.x * blockDim.x + threadIdx.x;
    if (g >= N_GRAPHS) return;
    float l0 = b[0], l1 = b[1];
#pragma unroll 4
    for (int k = 0; k < HID; ++k) {
        float v = s[g * HID + k];
        l0 += v * W[k * N_CLASS + 0];
        l1 += v * W[k * N_CLASS + 1];
    }
    float m   = fmaxf(l0, l1);
    float lse = m + logf(expf(l0 - m) + expf(l1 - m));
    out[g * N_CLASS + 0] = l0 - lse;
    out[g * N_CLASS + 1] = l1 - lse;
}

// ---------------------------------------------------------------------------
extern "C" void kernel_launch(void* const* d_in, const int* in_sizes, int n_in,
                              void* d_out, int out_size, void* d_ws, size_t ws_size,
                              hipStream_t stream) {
    (void)in_sizes; (void)n_in; (void)out_size; (void)ws_size;

    const float* x      = (const float*)d_in[0];
    const float* W1     = (const float*)d_in[1];
    const float* b1     = (const float*)d_in[2];
    const float* W2     = (const float*)d_in[3];
    const float* b2     = (const float*)d_in[4];
    const float* W3     = (const float*)d_in[5];
    const float* b3     = (const float*)d_in[6];
    const float* lin1_W = (const float*)d_in[7];
    const float* lin1_b = (const float*)d_in[8];
    const float* lin2_W = (const float*)d_in[9];
    const float* lin2_b = (const float*)d_in[10];
    const int*   eidx   = (const int*)d_in[11];   // [2, E] row-major
    const int*   e_src  = eidx;
    const int*   e_dst  = eidx + N_EDGES;
    float*       out    = (float*)d_out;          // [64, 2] log-probs

    // workspace layout (floats)
    float* ws    = (float*)d_ws;
    float* dinv  = ws;                                   // N       (deg -> dinv)
    float* enorm = dinv + 50048;                         // E
    float* h     = enorm + N_EDGES;                      // N*HID   (pre-agg GEMM out)
    float* hcat  = h + (size_t)N_NODES * HID;            // N*JK
    float* gfe   = hcat + (size_t)N_NODES * JK;          // 64*576
    float* gt    = gfe + N_GRAPHS * GFEAT;               // 64*64
    float* sbuf  = gt + N_GRAPHS * HID;                  // 64*64

    const int B = 256;
    auto cdiv = [](long long a, long long b) { return (int)((a + b - 1) / b); };

    // --- normalization coefficients ---
    zero_f32<<<cdiv(N_NODES, B), B, 0, stream>>>(dinv, N_NODES);
    deg_accum<<<cdiv(N_EDGES, B), B, 0, stream>>>(e_dst, dinv, N_EDGES);
    make_dinv<<<cdiv(N_NODES, B), B, 0, stream>>>(dinv, N_NODES);
    make_enorm<<<cdiv(N_EDGES, B), B, 0, stream>>>(e_src, e_dst, dinv, enorm, N_EDGES);

    // --- 3 GCN layers, outputs written into hcat slices (JK concat for free) ---
    const float* Ws[3] = {W1, W2, W3};
    const float* bs[3] = {b1, b2, b3};
    for (int l = 0; l < 3; ++l) {
        const float* Ain = (l == 0) ? x : (hcat + (l - 1) * HID);
        const int K   = (l == 0) ? F_IN : HID;
        const int lda = (l == 0) ? F_IN : JK;
        float* slice  = hcat + l * HID;  // stride JK

        gemm_wmma_f32<<<N_NODES / 16, 128, 0, stream>>>(Ain, Ws[l], h, N_NODES, K, lda, HID);
        init_out<<<cdiv((long long)N_NODES * HID, B), B, 0, stream>>>(h, dinv, bs[l], slice);
        scatter_edges<<<cdiv((long long)N_EDGES * HID, B), B, 0, stream>>>(e_src, e_dst, enorm, h, slice);
        relu_slice<<<cdiv((long long)N_NODES * HID, B), B, 0, stream>>>(slice);
    }

    // --- pooling: [64, 576] ---
    pool_kernel<<<N_GRAPHS, JK, 0, stream>>>(hcat, gfe);

    // --- head: lin1 (WMMA), bias+relu, lin2 + log_softmax ---
    gemm_wmma_f32<<<N_GRAPHS / 16, 128, 0, stream>>>(gfe, lin1_W, gt, N_GRAPHS, GFEAT, GFEAT, HID);
    bias_relu<<<cdiv(N_GRAPHS * HID, B), B, 0, stream>>>(gt, lin1_b, sbuf, N_GRAPHS * HID);
    head_kernel<<<1, 64, 0, stream>>>(sbuf, lin2_W, lin2_b, out);
}